// WQLinear_GEMM_14748917694674
// MI455X (gfx1250) — compile-verified
//
#include <hip/hip_runtime.h>

typedef float v2f __attribute__((ext_vector_type(2)));
typedef float v4f __attribute__((ext_vector_type(4)));
typedef float v8f __attribute__((ext_vector_type(8)));

#define M_DIM   16
#define K_DIM   4096
#define N_DIM   11008
#define N8      (N_DIM / 8)      // 1376 packed dwords per k-row
#define GROUPK  128
#define WAVES   8
#define KPW     (K_DIM / WAVES)  // 512 k per wave (multiple of GROUPK)
#define KSTEP   16               // k rows staged per wave-load

__global__ __launch_bounds__(WAVES * 32)
void awq_wmma_f32_kernel(const float* __restrict__ x,
                         const int*   __restrict__ qweight,
                         const int*   __restrict__ qzeros,
                         const float* __restrict__ scales,
                         const float* __restrict__ bias,
                         float* __restrict__ out)
{
    __shared__ float red[WAVES][32][8];   // 8 KB partial-C reduction buffer

    const int lane  = threadIdx.x & 31;
    const int wave  = threadIdx.x >> 5;
    const int tile  = blockIdx.x;               // one 16-column N tile per block
    const int n_base    = tile * 16;
    const int qcol_base = tile * 2;             // 2 packed dwords per 16 columns

    // ---- per-lane constants -------------------------------------------------
    const int nn    = lane & 15;                // B column n / A row m / C column
    const int kh    = lane >> 4;                // lane-half selector (ISA layout)
    const int mycol = (nn >> 3) & 1;            // which packed dword my column is in
    const int n8    = nn & 7;
    // AWQ_REVERSE_ORDER[j] = (j>>1) + 4*(j&1)  -> nibble shift for logical col
    const int shift = 4 * ((n8 >> 1) + 4 * (n8 & 1));

    const int kstart = wave * KPW;              // this wave's K slice

    // qweight staging: lane (kl<<1)|c loads dword qweight[k0+kl][qcol_base+c]
    const int qrow = lane >> 1;
    const int qc   = lane & 1;
    const int* qw_ptr = qweight + (size_t)(kstart + qrow) * N8 + qcol_base + qc;

    // k remap: hardware slots of wmma step j take global k = {2j, 2j+1, 8+2j, 8+2j+1}.
    // Lane-half kh therefore consumes contiguous k offsets [8*kh, 8*kh+8) -> b128 x2.
    const float* xrow = x + (size_t)nn * K_DIM + kstart + 8 * kh;

    v8f acc = {};                               // C/D: 16x16 f32 tile (8 VGPRs)
    float sc  = 0.0f;                           // group scale for my column
    float nzs = 0.0f;                           // -zero*scale for my column

    for (int k0 = 0; k0 < KPW; k0 += KSTEP) {
        // refresh group scale/zero every 128 k
        if (((kstart + k0) & (GROUPK - 1)) == 0) {
            const int g = (kstart + k0) >> 7;
            sc = scales[(size_t)g * N_DIM + n_base + nn];
            const int zdw = qzeros[(size_t)g * N8 + qcol_base + mycol];
            nzs = -(float)((zdw >> shift) & 15) * sc;
        }

        const int myqw = *qw_ptr;               // 16k x 16n nibble block across wave
        qw_ptr += KSTEP * N8;
        __builtin_prefetch(qw_ptr, 0, 0);       // global_prefetch_b8 for next block

        // two contiguous 16B activation loads cover this lane-half's 8 k values
        const v4f xa = *(const v4f*)(xrow + k0);
        const v4f xb = *(const v4f*)(xrow + k0 + 4);
        const float xk[8] = { xa.x, xa.y, xa.z, xa.w, xb.x, xb.y, xb.z, xb.w };

        #pragma unroll
        for (int j = 0; j < 4; ++j) {           // 4 x (16x16x4) WMMA per staged block
            const int klocal = 8 * kh + 2 * j;  // this lane-half's global k offset
            const int s0 = ((klocal + 0) << 1) | mycol;
            const int s1 = ((klocal + 1) << 1) | mycol;
            // lane-permute to fetch the packed dwords for my column's k values
            const int d0 = __builtin_amdgcn_ds_bpermute(s0 << 2, myqw);
            const int d1 = __builtin_amdgcn_ds_bpermute(s1 << 2, myqw);

            v2f b;                              // B slots: k = {.., ..} per half
            b.x = fmaf((float)((d0 >> shift) & 15), sc, nzs);
            b.y = fmaf((float)((d1 >> shift) & 15), sc, nzs);

            v2f a;                              // A slots: same k mapping as B
            a.x = xk[2 * j];
            a.y = xk[2 * j + 1];

            acc = __builtin_amdgcn_wmma_f32_16x16x4_f32(
                      /*neg_a=*/false, a, /*neg_b=*/false, b,
                      /*c_mod=*/(short)0, acc,
                      /*reuse_a=*/false, /*reuse_b=*/false);
        }
    }

    // ---- cross-wave K reduction via LDS ------------------------------------
    #pragma unroll
    for (int v = 0; v < 8; ++v) red[wave][lane][v] = acc[v];
    __syncthreads();

    if (wave == 0) {
        const float bv = bias[n_base + nn];
        #pragma unroll
        for (int v = 0; v < 8; ++v) {
            float s = 0.0f;
            #pragma unroll
            for (int w = 0; w < WAVES; ++w) s += red[w][lane][v];
            const int m = v + 8 * kh;           // C/D layout: row = v + 8*(lane>>4)
            out[(size_t)m * N_DIM + n_base + nn] = s + bv;
        }
    }
}

extern "C" void kernel_launch(void* const* d_in, const int* in_sizes, int n_in,
                              void* d_out, int out_size, void* d_ws, size_t ws_size,
                              hipStream_t stream) {
    (void)in_sizes; (void)n_in; (void)d_ws; (void)ws_size; (void)out_size;
    const float* x       = (const float*)d_in[0];
    const int*   qweight = (const int*)  d_in[1];
    const int*   qzeros  = (const int*)  d_in[2];
    const float* scales  = (const float*)d_in[3];
    const float* bias    = (const float*)d_in[4];
    float*       out     = (float*)d_out;

    dim3 grid(N_DIM / 16);        // 688 N-tiles
    dim3 block(WAVES * 32);       // 8 waves, wave32
    awq_wmma_f32_kernel<<<grid, block, 0, stream>>>(x, qweight, qzeros, scales, bias, out);
}